// _MSA3_17111149707929
// MI455X (gfx1250) — compile-verified
//
#include <hip/hip_runtime.h>
#include <hip/hip_bf16.h>
#include <math.h>

#define B_    4
#define NHEAD_ 16
#define L_    512
#define HD_   64
#define AL_   32
#define NHID_ 1024
#define C_    (NHEAD_*HD_)   // 1024

typedef __attribute__((ext_vector_type(2))) float v2f;
typedef __attribute__((ext_vector_type(8))) float v8f;

// ---------------------------------------------------------------------------
// Kernel 1: per-(b,h,l) attention with the collapsed axial term.
//   scores: s_u = (1/8) sum_d q[b,h,8d+l/64,l%64] * k[b,h, l+(3d+u)/64-1, (3d+u)%64]
//           s_ax = (1/8) sum_d q[...] * ax_k[b,h, d/2, l%64]   (x32 weight in softmax)
//   att[(b*L+l)*C + h*64+d] written K-contiguous for the GEMM.
// ---------------------------------------------------------------------------
__global__ __launch_bounds__(256) void msa_attn_kernel(
    const float* __restrict__ q,  const float* __restrict__ k,
    const float* __restrict__ v,  const float* __restrict__ axk,
    const float* __restrict__ axv, float* __restrict__ att)
{
    int idx = blockIdx.x * blockDim.x + threadIdx.x;      // (b*16+h)*512 + l
    if (idx >= B_ * NHEAD_ * L_) return;
    int l  = idx & (L_ - 1);
    int bh = idx >> 9;                                    // b*16 + h
    int lc = l & 63;                                      // l % 64
    int lr = l >> 6;                                      // l / 64

    const float* qbh  = q   + (size_t)bh * (L_ * HD_);
    const float* kbh  = k   + (size_t)bh * (L_ * HD_);
    const float* vbh  = v   + (size_t)bh * (L_ * HD_);
    const float* akbh = axk + (size_t)bh * (AL_ * HD_);
    const float* avbh = axv + (size_t)bh * (AL_ * HD_);

    float s0 = 0.f, s1 = 0.f, s2 = 0.f, sax = 0.f;
    #pragma unroll 4
    for (int d = 0; d < HD_; ++d) {
        float qd = qbh[(d * 8 + lr) * HD_ + lc];
        sax += qd * akbh[(d >> 1) * HD_ + lc];
        #pragma unroll
        for (int u = 0; u < 3; ++u) {
            int t   = 3 * d + u;
            int row = l + (t >> 6) - 1;                   // zero-padded window
            int c   = t & 63;
            float kv = (row >= 0 && row < L_) ? kbh[row * HD_ + c] : 0.f;
            if      (u == 0) s0 += qd * kv;
            else if (u == 1) s1 += qd * kv;
            else             s2 += qd * kv;
        }
    }
    const float scale = 0.125f;                           // 1/sqrt(64)
    s0 *= scale; s1 *= scale; s2 *= scale; sax *= scale;
    float mx  = fmaxf(fmaxf(s0, s1), fmaxf(s2, sax));
    float e0  = expf(s0 - mx), e1 = expf(s1 - mx), e2 = expf(s2 - mx);
    float eax = 32.f * expf(sax - mx);                    // 32 identical axial slots
    float inv = 1.f / (e0 + e1 + e2 + eax);
    e0 *= inv; e1 *= inv; e2 *= inv; eax *= inv;

    float* attp = att + ((size_t)((bh >> 4) * L_ + l)) * C_ + (bh & 15) * HD_;
    #pragma unroll 4
    for (int d = 0; d < HD_; ++d) {
        float acc = eax * avbh[(d >> 1) * HD_ + lc];
        #pragma unroll
        for (int u = 0; u < 3; ++u) {
            int t   = 3 * d + u;
            int row = l + (t >> 6) - 1;
            int c   = t & 63;
            float vv = (row >= 0 && row < L_) ? vbh[row * HD_ + c] : 0.f;
            acc += ((u == 0) ? e0 : (u == 1) ? e1 : e2) * vv;
        }
        attp[d] = acc;
    }
}

// ---------------------------------------------------------------------------
// Kernel 2: D[m,n] = sum_k att[m,k] * WO[k,n] + bias[n],  M=2048, N=K=1024.
// FP32 WMMA 16x16x4, one 16x16 tile per wave, 8 waves/block.
// A layout (ISA 7.12.2): lanes 0-15 -> M=lane, K={0,1} in v0/v1;
//                        lanes 16-31 -> M=lane-16, K={2,3}.
// B mirrored: lanes 0-15 -> N=lane, K={0,1}; lanes 16-31 -> K={2,3}.
// C/D: vgpr r -> M = r (lanes 0-15) / r+8 (lanes 16-31), N = lane%16.
// Store folds the [B,L,N] -> [B,N,L] transpose.
// ---------------------------------------------------------------------------
__global__ __launch_bounds__(256) void msa_wo_gemm_kernel(
    const float* __restrict__ att, const float* __restrict__ wo,
    const float* __restrict__ bias, float* __restrict__ out)
{
    int lane = threadIdx.x & 31;
    int wave = threadIdx.x >> 5;
    int m0   = blockIdx.x * 16;
    int n0   = (blockIdx.y * 8 + wave) * 16;
    int ln16 = lane & 15;
    int kSel = (lane < 16) ? 0 : 2;

    const float* arow = att + (size_t)(m0 + ln16) * C_;   // K-contiguous row
    const float* bcol = wo + n0 + ln16;                   // column of WO

    v8f acc = {0.f, 0.f, 0.f, 0.f, 0.f, 0.f, 0.f, 0.f};
    #pragma unroll 4
    for (int kk = 0; kk < C_; kk += 4) {
        v2f a = *(const v2f*)(arow + kk + kSel);          // 8B-aligned pair
        v2f b;
        b.x = bcol[(size_t)(kk + kSel)     * NHID_];
        b.y = bcol[(size_t)(kk + kSel + 1) * NHID_];
        acc = __builtin_amdgcn_wmma_f32_16x16x4_f32(false, a, false, b,
                                                    (short)0, acc, false, false);
    }

    int n  = n0 + ln16;
    float bn = bias[n];
    int mbase = m0 + ((lane < 16) ? 0 : 8);
    #pragma unroll
    for (int r = 0; r < 8; ++r) {
        int mm = mbase + r;
        int bb = mm >> 9;                                 // m / L
        int ll = mm & (L_ - 1);                           // m % L
        out[((size_t)(bb * NHID_ + n)) * L_ + ll] = acc[r] + bn;
    }
}

extern "C" void kernel_launch(void* const* d_in, const int* in_sizes, int n_in,
                              void* d_out, int out_size, void* d_ws, size_t ws_size,
                              hipStream_t stream) {
    const float* q    = (const float*)d_in[0];
    const float* k    = (const float*)d_in[1];
    const float* v    = (const float*)d_in[2];
    const float* axk  = (const float*)d_in[3];
    const float* axv  = (const float*)d_in[4];
    const float* wo   = (const float*)d_in[5];
    const float* bias = (const float*)d_in[6];
    float* out = (float*)d_out;
    float* att = (float*)d_ws;                            // B*L*C floats = 8 MB

    // Kernel 1: 32768 (b,h,l) threads
    msa_attn_kernel<<<dim3((B_ * NHEAD_ * L_) / 256), 256, 0, stream>>>(
        q, k, v, axk, axv, att);

    // Kernel 2: M/16 = 128 tile-rows, N covered by 8 blocks of 8 waves
    msa_wo_gemm_kernel<<<dim3((B_ * L_) / 16, NHID_ / (16 * 8)), 256, 0, stream>>>(
        att, wo, bias, out);
}